// EdgeNodeMessagePassing_27496380629763
// MI455X (gfx1250) — compile-verified
//
#include <hip/hip_runtime.h>

// ---------------------------------------------------------------------------
// GNN message passing on MI455X (gfx1250), fp32 path via V_WMMA_F32_16X16X4_F32
// LDS staging uses GLOBAL_LOAD_ASYNC_TO_LDS_B128 (ASYNCcnt) when available.
// ---------------------------------------------------------------------------

typedef __attribute__((ext_vector_type(2))) float v2f;
typedef __attribute__((ext_vector_type(8))) float v8f;
typedef __attribute__((ext_vector_type(4))) int v4i;

#define HDIM 128

#if __has_builtin(__builtin_amdgcn_global_load_async_to_lds_b128) && \
    __has_builtin(__builtin_amdgcn_s_wait_asynccnt)
#define USE_ASYNC_LDS 1
#else
#define USE_ASYNC_LDS 0
#endif

#if USE_ASYNC_LDS
typedef __attribute__((address_space(1))) v4i* g_v4i_p;
typedef __attribute__((address_space(3))) v4i* l_v4i_p;
__device__ __forceinline__ void async_b128(const float* g, float* l) {
  __builtin_amdgcn_global_load_async_to_lds_b128((g_v4i_p)g, (l_v4i_p)l, 0, 0);
}
__device__ __forceinline__ void async_wait_all() {
  __builtin_amdgcn_s_wait_asynccnt(0);
}
#endif

// Copy 16B global -> LDS (async path if available)
__device__ __forceinline__ void stage16(const float* g, float* l) {
#if USE_ASYNC_LDS
  async_b128(g, l);
#else
  *reinterpret_cast<float4*>(l) = *reinterpret_cast<const float4*>(g);
#endif
}

__device__ __forceinline__ void stage_fence() {
#if USE_ASYNC_LDS
  async_wait_all();
#endif
  __syncthreads();
}

// --- WMMA helpers (layouts per CDNA5 ISA 7.12.2) ---------------------------

__device__ __forceinline__ v8f wmma4(v2f a, v2f b, v8f c) {
  // (neg_a, A, neg_b, B, c_mod, C, reuse_a, reuse_b)
  return __builtin_amdgcn_wmma_f32_16x16x4_f32(false, a, false, b, (short)0, c,
                                               false, false);
}

// A 16x4 (MxK): lanes 0-15 -> M=lane, K=k0+{0,1}; lanes 16-31 -> K=k0+{2,3}
__device__ __forceinline__ v2f load_a_frag(const float* sA, int lane, int row0,
                                           int k0, int lda) {
  int r = row0 + (lane & 15);
  int k = k0 + ((lane >> 4) << 1);
  return *reinterpret_cast<const v2f*>(sA + r * lda + k);  // 8B aligned
}

// B 4x16 (KxN): VGPR v -> rows {v, v+2} split across lane halves
__device__ __forceinline__ v2f load_b_frag(const float* sB, int lane, int k0,
                                           int n0, int ldb) {
  int n = n0 + (lane & 15);
  int k = k0 + ((lane >> 4) << 1);
  v2f b;
  b.x = sB[k * ldb + n];
  b.y = sB[(k + 1) * ldb + n];
  return b;
}

// --- LDS layouts ------------------------------------------------------------
// A tiles: 64 rows, stride 132 (528B rows: 16B aligned; bank shift 4/row ->
//          conflict-free 16-row b64 A-frag fetches)
// B tiles: 128 rows, stride 144 (576B rows: 16B aligned; bank shift 16/row ->
//          lane halves hit disjoint banks)
constexpr int LDA = 132;
constexpr int LDB = 144;

constexpr int K1_SA = 0;
constexpr int K1_SB = 64 * LDA;
constexpr int K1_FLOATS = 64 * LDA + 128 * LDB;

constexpr int SA_OFF = 0;                      // 64*132 A tile (residual kept)
constexpr int SH_OFF = SA_OFF + 64 * LDA;      // 64*132 hidden activations
constexpr int SB_OFF = SH_OFF + 64 * LDA;      // 128*144 weight tile
constexpr int RED_OFF = SB_OFF + 128 * LDB;    // 512 LN partials
constexpr int MEAN_OFF = RED_OFF + 512;        // 64
constexpr int INV_OFF = MEAN_OFF + 64;         // 64
constexpr int ALPHA_OFF = INV_OFF + 64;        // 64
constexpr int G_OFF = ALPHA_OFF + 64;          // 128 gamma
constexpr int BT_OFF = G_OFF + 128;            // 128 beta
constexpr int ROWI_OFF = BT_OFF + 128;         // 64 (as int)
constexpr int COLI_OFF = ROWI_OFF + 64;        // 64 (as int)
constexpr int K3_FLOATS = COLI_OFF + 64;       // ~142 KB (<320KB/WGP)

// ===========================================================================
// Kernel 1: q = node_feat @ a_w + a_b     [N,128] x [128,128]
// ===========================================================================
__global__ __launch_bounds__(256) void qgemm_kernel(
    const float* __restrict__ X, const float* __restrict__ W,
    const float* __restrict__ bias, float* __restrict__ Q, int nrows) {
  extern __shared__ float smem[];
  float* sA = smem + K1_SA;
  float* sB = smem + K1_SB;

  const int t = threadIdx.x, wave = t >> 5, lane = t & 31;
  const int rbase = blockIdx.x * 64;

#pragma unroll
  for (int i = 0; i < 8; ++i) {  // 64 x 128 A tile
    int f = t + i * 256, r = f >> 5, c4 = (f & 31) << 2;
    int gr = rbase + r;
    if (gr >= nrows) gr = 0;
    stage16(X + (size_t)gr * HDIM + c4, &sA[r * LDA + c4]);
  }
#pragma unroll
  for (int i = 0; i < 16; ++i) {  // 128 x 128 W tile
    int f = t + i * 256, r = f >> 5, c4 = (f & 31) << 2;
    stage16(W + (size_t)r * HDIM + c4, &sB[r * LDB + c4]);
  }
  stage_fence();

  v8f acc[4] = {{}, {}, {}, {}};
  const int n0 = wave * 16;
  for (int k0 = 0; k0 < HDIM; k0 += 4) {
    v2f b = load_b_frag(sB, lane, k0, n0, LDB);
#pragma unroll
    for (int mt = 0; mt < 4; ++mt) {
      v2f a = load_a_frag(sA, lane, mt * 16, k0, LDA);
      acc[mt] = wmma4(a, b, acc[mt]);
    }
  }
  __syncthreads();  // everyone done reading sA/sB

  // dump accumulators (+bias) into sA, then coalesced float4 stores
  {
    const int col = n0 + (lane & 15);
    const float bv = bias[col];
    const int rowoff = (lane >> 4) << 3;
#pragma unroll
    for (int mt = 0; mt < 4; ++mt)
#pragma unroll
      for (int r = 0; r < 8; ++r)
        sA[(mt * 16 + r + rowoff) * LDA + col] = acc[mt][r] + bv;
  }
  __syncthreads();
  {
    const int row = t >> 2, q = t & 3;
    const int gr = rbase + row;
    if (gr < nrows) {
      float4* dst = reinterpret_cast<float4*>(Q + (size_t)gr * HDIM + q * 32);
      const float* src = &sA[row * LDA + q * 32];
#pragma unroll
      for (int i = 0; i < 8; ++i)
        dst[i] = *reinterpret_cast<const float4*>(src + i * 4);
    }
  }
}

// ===========================================================================
// Kernel 2: s[e] = leaky_relu(dot(q[row],q[col])); denom[col] += s[e]
// One wave32 per edge.
// ===========================================================================
__global__ __launch_bounds__(256) void attn_kernel(
    const float* __restrict__ Q, const int* __restrict__ ei,
    float* __restrict__ s_out, float* __restrict__ denom, int E) {
  const int t = threadIdx.x, wave = t >> 5, lane = t & 31;
  const long e = (long)blockIdx.x * 8 + wave;
  if (e >= E) return;  // wave-uniform
  const int row = ei[e];
  const int col = ei[(long)E + e];
  const float4 a =
      *reinterpret_cast<const float4*>(Q + (size_t)row * HDIM + lane * 4);
  const float4 b =
      *reinterpret_cast<const float4*>(Q + (size_t)col * HDIM + lane * 4);
  float d = a.x * b.x + a.y * b.y + a.z * b.z + a.w * b.w;
#pragma unroll
  for (int off = 16; off > 0; off >>= 1) d += __shfl_xor(d, off, 32);
  if (lane == 0) {
    float sv = d >= 0.f ? d : 0.01f * d;
    s_out[e] = sv;
    atomicAdd(&denom[col], sv);
  }
}

// ===========================================================================
// Kernel 3: edge MLP + LayerNorm + fused message scatter-add
// ===========================================================================
__global__ __launch_bounds__(256) void edge_mlp_kernel(
    const float* __restrict__ node_feat, const float* __restrict__ edge_feat,
    const int* __restrict__ ei, const float* __restrict__ w1,
    const float* __restrict__ b1, const float* __restrict__ w2,
    const float* __restrict__ b2, const float* __restrict__ gamma,
    const float* __restrict__ beta, const float* __restrict__ s_in,
    const float* __restrict__ denom, float* __restrict__ aggr,
    float* __restrict__ out_edge, int E) {
  extern __shared__ float smem[];
  float* sA = smem + SA_OFF;
  float* sH = smem + SH_OFF;
  float* sB = smem + SB_OFF;
  float* red = smem + RED_OFF;
  float* sMean = smem + MEAN_OFF;
  float* sInv = smem + INV_OFF;
  float* sAlpha = smem + ALPHA_OFF;
  float* sG = smem + G_OFF;
  float* sBt = smem + BT_OFF;
  int* sRowI = reinterpret_cast<int*>(smem + ROWI_OFF);
  int* sColI = reinterpret_cast<int*>(smem + COLI_OFF);

  const int t = threadIdx.x, wave = t >> 5, lane = t & 31;
  const long eBase = (long)blockIdx.x * 64;

  if (t < 64) {
    long e = eBase + t;
    bool valid = e < E;
    int r = valid ? ei[e] : 0;
    int c = valid ? ei[(long)E + e] : 0;
    sRowI[t] = r;
    sColI[t] = c;
    float al = 0.f;
    if (valid) {
      float dn = denom[c];
      dn = dn > 1e-6f ? dn : 1e-6f;
      al = s_in[e] / dn;
    }
    sAlpha[t] = al;
  } else if (t < 192) {
    int i = t - 64;
    sG[i] = gamma[i];
    sBt[i] = beta[i];
  }
  __syncthreads();

  v8f acc[4] = {{}, {}, {}, {}};
  const int n0 = wave * 16;

  // GEMM1: [xi | xj | edge_feat] @ e_w1 over 3 K-chunks of 128.
  // edge_feat is the LAST chunk so sA still holds it for the residual output.
  for (int chunk = 0; chunk < 3; ++chunk) {
#pragma unroll
    for (int i = 0; i < 8; ++i) {
      int f = t + i * 256, r = f >> 5, c4 = (f & 31) << 2;
      const float* src;
      if (chunk == 0)
        src = node_feat + (size_t)sRowI[r] * HDIM;
      else if (chunk == 1)
        src = node_feat + (size_t)sColI[r] * HDIM;
      else {
        long e = eBase + r;
        if (e >= E) e = 0;
        src = edge_feat + (size_t)e * HDIM;
      }
      stage16(src + c4, &sA[r * LDA + c4]);
    }
    const float* wsrc = w1 + (size_t)chunk * HDIM * HDIM;
#pragma unroll
    for (int i = 0; i < 16; ++i) {
      int f = t + i * 256, r = f >> 5, c4 = (f & 31) << 2;
      stage16(wsrc + (size_t)r * HDIM + c4, &sB[r * LDB + c4]);
    }
    stage_fence();
    for (int k0 = 0; k0 < HDIM; k0 += 4) {
      v2f b = load_b_frag(sB, lane, k0, n0, LDB);
#pragma unroll
      for (int mt = 0; mt < 4; ++mt) {
        v2f a = load_a_frag(sA, lane, mt * 16, k0, LDA);
        acc[mt] = wmma4(a, b, acc[mt]);
      }
    }
    __syncthreads();
  }

  // bias + relu, dump h1 -> sH; concurrently stage e_w2 into sB
  {
    const int col = n0 + (lane & 15);
    const float bv = b1[col];
    const int rowoff = (lane >> 4) << 3;
#pragma unroll
    for (int mt = 0; mt < 4; ++mt)
#pragma unroll
      for (int r = 0; r < 8; ++r) {
        float v = acc[mt][r] + bv;
        sH[(mt * 16 + r + rowoff) * LDA + col] = v > 0.f ? v : 0.f;
      }
  }
#pragma unroll
  for (int i = 0; i < 16; ++i) {
    int f = t + i * 256, r = f >> 5, c4 = (f & 31) << 2;
    stage16(w2 + (size_t)r * HDIM + c4, &sB[r * LDB + c4]);
  }
  stage_fence();

  // GEMM2: h1 @ e_w2
  v8f acc2[4] = {{}, {}, {}, {}};
  for (int k0 = 0; k0 < HDIM; k0 += 4) {
    v2f b = load_b_frag(sB, lane, k0, n0, LDB);
#pragma unroll
    for (int mt = 0; mt < 4; ++mt) {
      v2f a = load_a_frag(sH, lane, mt * 16, k0, LDA);
      acc2[mt] = wmma4(a, b, acc2[mt]);
    }
  }
  __syncthreads();

  // bias + relu, dump h2 -> sH
  {
    const int col = n0 + (lane & 15);
    const float bv = b2[col];
    const int rowoff = (lane >> 4) << 3;
#pragma unroll
    for (int mt = 0; mt < 4; ++mt)
#pragma unroll
      for (int r = 0; r < 8; ++r) {
        float v = acc2[mt][r] + bv;
        sH[(mt * 16 + r + rowoff) * LDA + col] = v > 0.f ? v : 0.f;
      }
  }
  __syncthreads();

  // LayerNorm: 4 threads per row, 32 elements each
  {
    const int row = t >> 2, q = t & 3;
    float sum = 0.f, sq = 0.f;
    const float* hp = &sH[row * LDA + q * 32];
#pragma unroll
    for (int i = 0; i < 32; ++i) {
      float v = hp[i];
      sum += v;
      sq += v * v;
    }
    red[row * 4 + q] = sum;
    red[256 + row * 4 + q] = sq;
  }
  __syncthreads();
  if (t < 64) {
    float sum = red[t * 4] + red[t * 4 + 1] + red[t * 4 + 2] + red[t * 4 + 3];
    float sq = red[256 + t * 4] + red[256 + t * 4 + 1] + red[256 + t * 4 + 2] +
               red[256 + t * 4 + 3];
    float m = sum * (1.f / 128.f);
    float var = sq * (1.f / 128.f) - m * m;
    sMean[t] = m;
    sInv[t] = rsqrtf(var + 1e-5f);
  }
  __syncthreads();

  // Epilogue: out_edge = LN(h2) + edge_feat (edge_feat still in sA);
  //           aggr[col] += alpha * LN(h2)  (f32 scatter-add)
  {
    const int row = t >> 2, q = t & 3;
    const long e = eBase + row;
    if (e < E) {
      const float m = sMean[row], inv = sInv[row], al = sAlpha[row];
      const int c = sColI[row];
      float* outp = out_edge + (size_t)e * HDIM;
      float* agp = aggr + (size_t)c * HDIM;
      const float* hp = &sH[row * LDA];
      const float* ap = &sA[row * LDA];
#pragma unroll
      for (int g4 = 0; g4 < 8; ++g4) {
        const int n = q * 32 + g4 * 4;
        float v0 = (hp[n + 0] - m) * inv * sG[n + 0] + sBt[n + 0];
        float v1 = (hp[n + 1] - m) * inv * sG[n + 1] + sBt[n + 1];
        float v2 = (hp[n + 2] - m) * inv * sG[n + 2] + sBt[n + 2];
        float v3 = (hp[n + 3] - m) * inv * sG[n + 3] + sBt[n + 3];
        float4 o;
        o.x = v0 + ap[n + 0];
        o.y = v1 + ap[n + 1];
        o.z = v2 + ap[n + 2];
        o.w = v3 + ap[n + 3];
        *reinterpret_cast<float4*>(outp + n) = o;
        atomicAdd(&agp[n + 0], v0 * al);
        atomicAdd(&agp[n + 1], v1 * al);
        atomicAdd(&agp[n + 2], v2 * al);
        atomicAdd(&agp[n + 3], v3 * al);
      }
    }
  }
}

// ===========================================================================
// Kernel 4: node MLP + LayerNorm + residual
// ===========================================================================
__global__ __launch_bounds__(256) void node_mlp_kernel(
    const float* __restrict__ node_feat, const float* __restrict__ aggr,
    const float* __restrict__ w1, const float* __restrict__ b1,
    const float* __restrict__ w2, const float* __restrict__ b2,
    const float* __restrict__ gamma, const float* __restrict__ beta,
    float* __restrict__ out_node, int N) {
  extern __shared__ float smem[];
  float* sA = smem + SA_OFF;
  float* sH = smem + SH_OFF;
  float* sB = smem + SB_OFF;
  float* red = smem + RED_OFF;
  float* sMean = smem + MEAN_OFF;
  float* sInv = smem + INV_OFF;
  float* sG = smem + G_OFF;
  float* sBt = smem + BT_OFF;

  const int t = threadIdx.x, wave = t >> 5, lane = t & 31;
  const int rbase = blockIdx.x * 64;

  if (t < 128) {
    sG[t] = gamma[t];
    sBt[t] = beta[t];
  }
  __syncthreads();

  v8f acc[4] = {{}, {}, {}, {}};
  const int n0 = wave * 16;

  // chunk 0: aggr vs n_w1 rows [128,256); chunk 1: node_feat vs rows [0,128)
  // node_feat last so sA holds it for the residual.
  for (int chunk = 0; chunk < 2; ++chunk) {
    const float* xsrc = (chunk == 0) ? aggr : node_feat;
    const float* wsrc = (chunk == 0) ? (w1 + (size_t)HDIM * HDIM) : w1;
#pragma unroll
    for (int i = 0; i < 8; ++i) {
      int f = t + i * 256, r = f >> 5, c4 = (f & 31) << 2;
      int gr = rbase + r;
      if (gr >= N) gr = 0;
      stage16(xsrc + (size_t)gr * HDIM + c4, &sA[r * LDA + c4]);
    }
#pragma unroll
    for (int i = 0; i < 16; ++i) {
      int f = t + i * 256, r = f >> 5, c4 = (f & 31) << 2;
      stage16(wsrc + (size_t)r * HDIM + c4, &sB[r * LDB + c4]);
    }
    stage_fence();
    for (int k0 = 0; k0 < HDIM; k0 += 4) {
      v2f b = load_b_frag(sB, lane, k0, n0, LDB);
#pragma unroll
      for (int mt = 0; mt < 4; ++mt) {
        v2f a = load_a_frag(sA, lane, mt * 16, k0, LDA);
        acc[mt] = wmma4(a, b, acc[mt]);
      }
    }
    __syncthreads();
  }

  {
    const int col = n0 + (lane & 15);
    const float bv = b1[col];
    const int rowoff = (lane >> 4) << 3;
#pragma unroll
    for (int mt = 0; mt < 4; ++mt)
#pragma unroll
      for (int r = 0; r < 8; ++r) {
        float v = acc[mt][r] + bv;
        sH[(mt * 16 + r + rowoff) * LDA + col] = v > 0.f ? v : 0.f;
      }
  }
#pragma unroll
  for (int i = 0; i < 16; ++i) {
    int f = t + i * 256, r = f >> 5, c4 = (f & 31) << 2;
    stage16(w2 + (size_t)r * HDIM + c4, &sB[r * LDB + c4]);
  }
  stage_fence();

  v8f acc2[4] = {{}, {}, {}, {}};
  for (int k0 = 0; k0 < HDIM; k0 += 4) {
    v2f b = load_b_frag(sB, lane, k0, n0, LDB);
#pragma unroll
    for (int mt = 0; mt < 4; ++mt) {
      v2f a = load_a_frag(sH, lane, mt * 16, k0, LDA);
      acc2[mt] = wmma4(a, b, acc2[mt]);
    }
  }
  __syncthreads();

  {
    const int col = n0 + (lane & 15);
    const float bv = b2[col];
    const int rowoff = (lane >> 4) << 3;
#pragma unroll
    for (int mt = 0; mt < 4; ++mt)
#pragma unroll
      for (int r = 0; r < 8; ++r) {
        float v = acc2[mt][r] + bv;
        sH[(mt * 16 + r + rowoff) * LDA + col] = v > 0.f ? v : 0.f;
      }
  }
  __syncthreads();

  {
    const int row = t >> 2, q = t & 3;
    float sum = 0.f, sq = 0.f;
    const float* hp = &sH[row * LDA + q * 32];
#pragma unroll
    for (int i = 0; i < 32; ++i) {
      float v = hp[i];
      sum += v;
      sq += v * v;
    }
    red[row * 4 + q] = sum;
    red[256 + row * 4 + q] = sq;
  }
  __syncthreads();
  if (t < 64) {
    float sum = red[t * 4] + red[t * 4 + 1] + red[t * 4 + 2] + red[t * 4 + 3];
    float sq = red[256 + t * 4] + red[256 + t * 4 + 1] + red[256 + t * 4 + 2] +
               red[256 + t * 4 + 3];
    float m = sum * (1.f / 128.f);
    float var = sq * (1.f / 128.f) - m * m;
    sMean[t] = m;
    sInv[t] = rsqrtf(var + 1e-5f);
  }
  __syncthreads();

  {
    const int row = t >> 2, q = t & 3;
    const int gr = rbase + row;
    if (gr < N) {
      const float m = sMean[row], inv = sInv[row];
      float* outp = out_node + (size_t)gr * HDIM;
      const float* hp = &sH[row * LDA];
      const float* ap = &sA[row * LDA];  // node_feat residual
#pragma unroll
      for (int g4 = 0; g4 < 8; ++g4) {
        const int n = q * 32 + g4 * 4;
        float4 o;
        o.x = (hp[n + 0] - m) * inv * sG[n + 0] + sBt[n + 0] + ap[n + 0];
        o.y = (hp[n + 1] - m) * inv * sG[n + 1] + sBt[n + 1] + ap[n + 1];
        o.z = (hp[n + 2] - m) * inv * sG[n + 2] + sBt[n + 2] + ap[n + 2];
        o.w = (hp[n + 3] - m) * inv * sG[n + 3] + sBt[n + 3] + ap[n + 3];
        *reinterpret_cast<float4*>(outp + n) = o;
      }
    }
  }
}

// ===========================================================================
extern "C" void kernel_launch(void* const* d_in, const int* in_sizes, int n_in,
                              void* d_out, int out_size, void* d_ws,
                              size_t ws_size, hipStream_t stream) {
  (void)n_in;
  (void)out_size;
  (void)ws_size;
  const float* node_feat = (const float*)d_in[0];
  const float* edge_feat = (const float*)d_in[1];
  const int* edge_index = (const int*)d_in[2];
  const float* e_w1 = (const float*)d_in[3];
  const float* e_b1 = (const float*)d_in[4];
  const float* e_w2 = (const float*)d_in[5];
  const float* e_b2 = (const float*)d_in[6];
  const float* e_g = (const float*)d_in[7];
  const float* e_beta = (const float*)d_in[8];
  const float* n_w1 = (const float*)d_in[9];
  const float* n_b1 = (const float*)d_in[10];
  const float* n_w2 = (const float*)d_in[11];
  const float* n_b2 = (const float*)d_in[12];
  const float* n_g = (const float*)d_in[13];
  const float* n_beta = (const float*)d_in[14];
  const float* a_w = (const float*)d_in[15];
  const float* a_b = (const float*)d_in[16];

  const int N = in_sizes[0] / HDIM;
  const int E = in_sizes[1] / HDIM;

  // workspace: q[N*H] | s[E] | denom[N] | aggr[N*H]
  float* q = (float*)d_ws;
  float* s = q + (size_t)N * HDIM;
  float* denom = s + (size_t)E;
  float* aggr = denom + (size_t)N;

  float* out_node = (float*)d_out;
  float* out_edge = out_node + (size_t)N * HDIM;

  // zero the atomic accumulation buffers (denom and aggr are contiguous)
  (void)hipMemsetAsync(denom, 0,
                       sizeof(float) * ((size_t)N + (size_t)N * HDIM), stream);

  const size_t smem1 = sizeof(float) * (size_t)K1_FLOATS;
  const size_t smem3 = sizeof(float) * (size_t)K3_FLOATS;

  qgemm_kernel<<<(N + 63) / 64, 256, smem1, stream>>>(node_feat, a_w, a_b, q,
                                                      N);
  attn_kernel<<<(E + 7) / 8, 256, 0, stream>>>(q, edge_index, s, denom, E);
  edge_mlp_kernel<<<(E + 63) / 64, 256, smem3, stream>>>(
      node_feat, edge_feat, edge_index, e_w1, e_b1, e_w2, e_b2, e_g, e_beta, s,
      denom, aggr, out_edge, E);
  node_mlp_kernel<<<(N + 63) / 64, 256, smem3, stream>>>(
      node_feat, aggr, n_w1, n_b1, n_w2, n_b2, n_g, n_beta, out_node, N);
}